// SpatialGraphConv_41248865911146
// MI455X (gfx1250) — compile-verified
//
#include <hip/hip_runtime.h>

// ---------------------------------------------------------------------------
// SpatialGraphConv for MI455X (gfx1250, wave32)
//   nodes: (N,32) f32, edges: 1.6M with sorted receivers.
//   Edge phase: memory/atomic bound -> fp32 atomics into L2-resident per-node
//   accumulators.  Node epilogue: two 32x32 fp32 GEMMs -> V_WMMA_F32_16X16X4_F32.
// ---------------------------------------------------------------------------

#define K_IND   16
#define K_FEAT  32
#define DHID    64

typedef __attribute__((ext_vector_type(2))) float v2f;
typedef __attribute__((ext_vector_type(8))) float v8f;

// float atomic-max via monotone int/uint ordering trick (init = -inf bits)
__device__ __forceinline__ void atomic_fmax(float* addr, float v) {
  if (v >= 0.0f) atomicMax((int*)addr, __float_as_int(v));
  else           atomicMin((unsigned int*)addr, __float_as_uint(v));
}

// column k (0..15) of relu(d*w1+b1) @ w2 + b2  (w1:(1,64), w2:(64,16))
__device__ __forceinline__ float mlp_col(float d, int k,
                                         const float* __restrict__ w1,
                                         const float* __restrict__ b1,
                                         const float* __restrict__ w2,
                                         const float* __restrict__ b2) {
  float acc = b2[k];
#pragma unroll 8
  for (int h = 0; h < DHID; ++h) {
    float hv = fmaxf(fmaf(d, w1[h], b1[h]), 0.0f);
    acc = fmaf(hv, w2[h * K_IND + k], acc);
  }
  return acc;
}

// one-hot bin j such that j/16 < d < (j+1)/16 (strict, like heaviside(...,0))
__device__ __forceinline__ int edge_bin(float d, bool& valid) {
  float fb = d * (float)K_IND;        // DIST_MAX == 1.0
  int j = (int)floorf(fb);
  valid = (j >= 0) && (j < K_IND) && (fb > (float)j) && (fb < (float)(j + 1));
  return j;
}

// ---------------------------------------------------------------------------
// K0: zero/init workspace accumulators
// ---------------------------------------------------------------------------
__global__ void sgc_init_ws(float* __restrict__ ind_sums, float* __restrict__ mlp_max,
                            float* __restrict__ expsum,   float* __restrict__ gathered,
                            int n) {
  int i = blockIdx.x * blockDim.x + threadIdx.x;
  if (i < n * K_FEAT) gathered[i] = 0.0f;
  if (i < n * K_IND) {
    ind_sums[i] = 0.0f;
    expsum[i]   = 0.0f;
    mlp_max[i]  = __int_as_float(0xFF800000); // -inf
  }
}

// ---------------------------------------------------------------------------
// K1: per-edge, 16 lanes/edge (lane = column): indicator counts + segment max
// ---------------------------------------------------------------------------
__global__ void sgc_edge_binmax(const float* __restrict__ dist, const int* __restrict__ recv,
                                const float* __restrict__ w1, const float* __restrict__ b1,
                                const float* __restrict__ w2, const float* __restrict__ b2,
                                float* __restrict__ ind_sums, float* __restrict__ mlp_max,
                                int E) {
  int t = blockIdx.x * blockDim.x + threadIdx.x;
  int e = t >> 4, k = t & 15;
  if (e >= E) return;
  float d = dist[e];
  int   r = recv[e];
  bool valid; int j = edge_bin(d, valid);
  if (valid && k == j) atomicAdd(&ind_sums[r * K_IND + j], 1.0f);
  float m = mlp_col(d, k, w1, b1, w2, b2);
  atomic_fmax(&mlp_max[r * K_IND + k], m);
}

// ---------------------------------------------------------------------------
// K2: per-edge, 16 lanes/edge: softmax denominators
// ---------------------------------------------------------------------------
__global__ void sgc_edge_expsum(const float* __restrict__ dist, const int* __restrict__ recv,
                                const float* __restrict__ w1, const float* __restrict__ b1,
                                const float* __restrict__ w2, const float* __restrict__ b2,
                                const float* __restrict__ mlp_max, float* __restrict__ expsum,
                                int E) {
  int t = blockIdx.x * blockDim.x + threadIdx.x;
  int e = t >> 4, k = t & 15;
  if (e >= E) return;
  float d = dist[e];
  int   r = recv[e];
  float m = mlp_col(d, k, w1, b1, w2, b2);
  float ex = __expf(m - mlp_max[r * K_IND + k]);
  atomicAdd(&expsum[r * K_IND + k], ex);
}

// ---------------------------------------------------------------------------
// K3: per-edge, 32 lanes/edge (lane = feature): weights * |a*xr-(1-a)*xs|^b
//     scattered into gathered[recv].  Coalesced 128B row loads; prefetch a
//     future sender row into cache (gfx1250 global_prefetch path).
// ---------------------------------------------------------------------------
__global__ void sgc_edge_scatter(const float* __restrict__ nodes, const float* __restrict__ dist,
                                 const float* __restrict__ pad,
                                 const int* __restrict__ recv, const int* __restrict__ send,
                                 const float* __restrict__ w1, const float* __restrict__ b1,
                                 const float* __restrict__ w2, const float* __restrict__ b2,
                                 const float* __restrict__ av, const float* __restrict__ bv,
                                 const float* __restrict__ ind_sums,
                                 const float* __restrict__ mlp_max,
                                 const float* __restrict__ expsum,
                                 float* __restrict__ gathered, int E) {
  int t = blockIdx.x * blockDim.x + threadIdx.x;
  int e = t >> 5, j = t & 31;
  if (e >= E) return;
  int r = recv[e];
  int s = send[e];
  if (j == 0 && (e + 16) < E) {
    int sn = send[e + 16];
    __builtin_prefetch(nodes + (size_t)sn * K_FEAT, 0, 1);
  }
  float d = dist[e];
  float w;
  if (j < K_IND) {
    bool valid; int bj = edge_bin(d, valid);
    float ind = (valid && j == bj) ? 1.0f : 0.0f;
    w = ind / (ind_sums[r * K_IND + j] + 1e-5f);
  } else {
    int k = j - K_IND;
    float m = mlp_col(d, k, w1, b1, w2, b2);
    w = __expf(m - mlp_max[r * K_IND + k]) / expsum[r * K_IND + k];
  }
  w *= pad[e];
  float ac = fminf(fmaxf(av[0], 0.0f), 1.0f);
  float bc = fabsf(bv[0]);
  float diff = fabsf(ac * nodes[(size_t)r * K_FEAT + j] -
                     (1.0f - ac) * nodes[(size_t)s * K_FEAT + j]);
  float pd = __powf(diff, bc);   // diff==0 -> exp2(-inf)=0, matches 0^b for b>0
  atomicAdd(&gathered[(size_t)r * K_FEAT + j], w * pd);
}

// ---------------------------------------------------------------------------
// K4: out = relu(nodes @ wg_self + gathered @ wg_gath + bg)  via fp32 WMMA.
//     One wave -> 16-row x 32-col output tile: 2 GEMMs x 2 col-tiles x 8
//     k-chunks = 32 V_WMMA_F32_16X16X4_F32.  Bias preloaded into accumulator,
//     ReLU fused on store.  All branches are wave-uniform (EXEC all-ones).
// ---------------------------------------------------------------------------
__global__ void sgc_node_out_wmma(const float* __restrict__ nodes,
                                  const float* __restrict__ gathered,
                                  const float* __restrict__ wg_self,
                                  const float* __restrict__ wg_gath,
                                  const float* __restrict__ bg,
                                  float* __restrict__ out, int n) {
  int wave = (blockIdx.x * blockDim.x + threadIdx.x) >> 5;
  int lane = threadIdx.x & 31;
  int row0 = wave * 16;
  if (row0 >= n) return;                 // wave-uniform exit

  int lrow  = lane & 15;                 // output column within col-tile / A row
  int khalf = lane >> 4;                 // 0: K={0,1}; 1: K={2,3} per chunk

  // accumulators seeded with bias (C-matrix col = lrow for both lane halves)
  float c0 = bg[lrow], c1 = bg[K_IND + lrow];
  v8f acc0, acc1;
#pragma unroll
  for (int v = 0; v < 8; ++v) { acc0[v] = c0; acc1[v] = c1; }

  // ---- nodes @ wg_self ----
#pragma unroll
  for (int kc = 0; kc < 8; ++kc) {
    int kb = kc * 4 + 2 * khalf;
    v2f a = *(const v2f*)(nodes + (size_t)(row0 + lrow) * K_FEAT + kb);
    v2f b0, b1;
    b0.x = wg_self[(kb + 0) * K_FEAT + lrow];
    b0.y = wg_self[(kb + 1) * K_FEAT + lrow];
    b1.x = wg_self[(kb + 0) * K_FEAT + K_IND + lrow];
    b1.y = wg_self[(kb + 1) * K_FEAT + K_IND + lrow];
    acc0 = __builtin_amdgcn_wmma_f32_16x16x4_f32(false, a, false, b0, (short)0, acc0, false, false);
    acc1 = __builtin_amdgcn_wmma_f32_16x16x4_f32(false, a, false, b1, (short)0, acc1, false, false);
  }

  // ---- gathered @ wg_gath ----
#pragma unroll
  for (int kc = 0; kc < 8; ++kc) {
    int kb = kc * 4 + 2 * khalf;
    v2f a = *(const v2f*)(gathered + (size_t)(row0 + lrow) * K_FEAT + kb);
    v2f b0, b1;
    b0.x = wg_gath[(kb + 0) * K_FEAT + lrow];
    b0.y = wg_gath[(kb + 1) * K_FEAT + lrow];
    b1.x = wg_gath[(kb + 0) * K_FEAT + K_IND + lrow];
    b1.y = wg_gath[(kb + 1) * K_FEAT + K_IND + lrow];
    acc0 = __builtin_amdgcn_wmma_f32_16x16x4_f32(false, a, false, b0, (short)0, acc0, false, false);
    acc1 = __builtin_amdgcn_wmma_f32_16x16x4_f32(false, a, false, b1, (short)0, acc1, false, false);
  }

  // ---- ReLU + store (D layout: VGPR v -> rows v / v+8) ----
#pragma unroll
  for (int v = 0; v < 8; ++v) {
    int row = row0 + v + 8 * khalf;
    out[(size_t)row * K_FEAT + lrow]         = fmaxf(acc0[v], 0.0f);
    out[(size_t)row * K_FEAT + K_IND + lrow] = fmaxf(acc1[v], 0.0f);
  }
}

// ---------------------------------------------------------------------------
extern "C" void kernel_launch(void* const* d_in, const int* in_sizes, int n_in,
                              void* d_out, int out_size, void* d_ws, size_t ws_size,
                              hipStream_t stream) {
  const float* nodes    = (const float*)d_in[0];
  const float* distance = (const float*)d_in[1];
  const float* pad      = (const float*)d_in[2];
  const int*   recv     = (const int*)d_in[3];
  const int*   send     = (const int*)d_in[4];
  const float* w1       = (const float*)d_in[5];
  const float* b1       = (const float*)d_in[6];
  const float* w2       = (const float*)d_in[7];
  const float* b2       = (const float*)d_in[8];
  const float* av       = (const float*)d_in[9];
  const float* bv       = (const float*)d_in[10];
  const float* wg_self  = (const float*)d_in[11];
  const float* wg_gath  = (const float*)d_in[12];
  const float* bg       = (const float*)d_in[13];
  float* out = (float*)d_out;

  const int n = in_sizes[0] / K_FEAT;   // 100000
  const int E = in_sizes[1];            // 1600000

  // workspace layout (fp32): [ind_sums n*16][mlp_max n*16][expsum n*16][gathered n*32]
  float* ws       = (float*)d_ws;
  float* ind_sums = ws;
  float* mlp_max  = ws + (size_t)n * 16;
  float* expsum   = ws + (size_t)n * 32;
  float* gathered = ws + (size_t)n * 48;

  const int TB = 256;

  int init_threads = n * K_FEAT;
  sgc_init_ws<<<(init_threads + TB - 1) / TB, TB, 0, stream>>>(
      ind_sums, mlp_max, expsum, gathered, n);

  int e16 = E * 16;
  sgc_edge_binmax<<<(e16 + TB - 1) / TB, TB, 0, stream>>>(
      distance, recv, w1, b1, w2, b2, ind_sums, mlp_max, E);

  sgc_edge_expsum<<<(e16 + TB - 1) / TB, TB, 0, stream>>>(
      distance, recv, w1, b1, w2, b2, mlp_max, expsum, E);

  int e32 = E * 32;
  sgc_edge_scatter<<<(e32 + TB - 1) / TB, TB, 0, stream>>>(
      nodes, distance, pad, recv, send, w1, b1, w2, b2, av, bv,
      ind_sums, mlp_max, expsum, gathered, E);

  int waves = (n + 15) / 16;            // one wave per 16-row tile
  int out_threads = waves * 32;
  sgc_node_out_wmma<<<(out_threads + TB - 1) / TB, TB, 0, stream>>>(
      nodes, gathered, wg_self, wg_gath, bg, out, n);
}